// DiagonalTraining_89936615178631
// MI455X (gfx1250) — compile-verified
//
#include <hip/hip_runtime.h>
#include <hip/hip_bf16.h>

// Per-anti-diagonal batched Linear:
//   out[b, r, c] = sum_k W[r+c, r-r0, k] * x[b, r0+k, (r+c)-r0-k] + bias[r+c, r-r0]
// One workgroup per diagonal d (511 blocks). Memory-bound (~110-200MB @ 23.3TB/s
// => ~5-9us floor; 2.86 GFLOP of math), so keep fp32 and use V_WMMA_F32_16X16X4_F32.
// X staging uses the CDNA5 async gather path: GLOBAL_LOAD_ASYNC_TO_LDS_B32
// (per-lane global addr -> per-lane LDS addr, ASYNCcnt), no VGPR round-trip.

typedef __attribute__((ext_vector_type(2))) float v2f;
typedef __attribute__((ext_vector_type(8))) float v8f;

#define SDIM 256
#define BDIM 128
#define XPAD 2   // LDS row padding (floats) to break bank conflicts

__global__ __launch_bounds__(256)
void DiagonalTraining_wmma_kernel(const float* __restrict__ x,
                                  const float* __restrict__ W,
                                  const float* __restrict__ bias,
                                  float* __restrict__ out)
{
    extern __shared__ float Xs[];   // [BDIM][np + XPAD], np <= 256 -> max 132,096 B
                                    // (sole LDS allocation => LDS byte offset 0)

    const int d  = blockIdx.x;                       // diagonal id 0..510
    const int r0 = (d >= SDIM) ? (d - SDIM + 1) : 0; // first row on this diagonal
    const int n  = (d < SDIM) ? (d + 1) : (2 * SDIM - 1 - d); // diagonal length
    const int np = (n + 15) & ~15;                   // padded to WMMA tile multiple
    const int xrs = np + XPAD;                       // Xs row stride in floats
    const int diagBase = r0 * SDIM + (d - r0);       // offset of diag start in one image
    const int dstride  = SDIM - 1;                   // 255-float stride along the diagonal

    const int lane = threadIdx.x & 31;
    const int wv   = threadIdx.x >> 5;               // wave id 0..7

    // ---- Stage the strided diagonal gather into LDS via async tensor-path loads.
    // Each lane copies x[b, r0+k, d-r0-k] from global memory straight into Xs row b, column k (ASYNCcnt),
    // K padding is zero-filled with plain ds stores (DScnt, waited by barrier).
    for (int b = wv; b < BDIM; b += 8) {
        const float* xb = x + (size_t)b * SDIM * SDIM + diagBase;
        float* row = Xs + b * xrs;
        for (int k = lane; k < np; k += 32) {
            if (k < n) {
                const unsigned ldsoff = (unsigned)(b * xrs + k) * 4u; // byte offset in LDS
                const float* gp = xb + (size_t)k * dstride;
                asm volatile("global_load_async_to_lds_b32 %0, %1, off"
                             :: "v"(ldsoff), "v"(gp)
                             : "memory");
            } else {
                row[k] = 0.0f;
            }
        }
    }
    asm volatile("s_wait_asynccnt 0" ::: "memory");  // all async gathers landed in LDS
    __syncthreads();

    // ---- WMMA GEMM: Y(128 x np) = Xs(128 x np) * W[d]^T, per 16x16 output tile ----
    const int mrow  = lane & 15;   // A: M row / B: N col / C: N col
    const int khalf = lane >> 4;   // lane half selects K pair {0,1} vs {2,3}
    const float* Wd = W + (size_t)d * SDIM * SDIM;

    for (int j0 = 0; j0 < np; j0 += 16) {
        // Accumulator initialized with bias (bias depends only on the output column,
        // which is lane&15 for every one of the 8 C VGPRs).
        const float bv = bias[d * SDIM + j0 + mrow];
        v8f acc;
        #pragma unroll
        for (int v = 0; v < 8; ++v) acc[v] = bv;

        const float* arow = Xs + (wv * 16 + mrow) * xrs + khalf * 2;
        const float* brow = Wd + (j0 + mrow) * SDIM + khalf * 2;

        for (int k0 = 0; k0 < np; k0 += 4) {
            // A fragment: 16x4 fp32 tile of Xs (ds_load_2addr_b32 per lane)
            v2f a;  a.x  = arow[k0];  a.y  = arow[k0 + 1];
            // B fragment: B[k][j] = W[d, j0+j, k0+k]; W's memory zero-padding
            // makes partial tiles correct with no masking (global_load_b64 per lane).
            v2f bm; bm.x = brow[k0];  bm.y = brow[k0 + 1];
            // WGP-scope prefetch of the W stream one cacheline-group ahead
            // (cachelines ARE reused across adjacent K iterations by this wave).
            __builtin_prefetch(brow + k0 + 64, 0, 3);
            acc = __builtin_amdgcn_wmma_f32_16x16x4_f32(
                /*neg_a=*/false, a, /*neg_b=*/false, bm,
                /*c_mod=*/(short)0, acc, /*reuse_a=*/false, /*reuse_b=*/false);
        }

        // ---- Scatter the 16x16 tile back onto the diagonal ----
        const int m = j0 + mrow;                   // output position along diagonal
        if (m < n) {
            const size_t o = (size_t)diagBase + (size_t)m * dstride;
            #pragma unroll
            for (int v = 0; v < 8; ++v) {
                const int b = wv * 16 + khalf * 8 + v;   // C layout: M = v + 8*(lane>=16)
                out[(size_t)b * SDIM * SDIM + o] = acc[v];
            }
        }
    }
}

extern "C" void kernel_launch(void* const* d_in, const int* in_sizes, int n_in,
                              void* d_out, int out_size, void* d_ws, size_t ws_size,
                              hipStream_t stream) {
    const float* x    = (const float*)d_in[0];  // [128, 256, 256] fp32
    const float* W    = (const float*)d_in[1];  // [511, 256, 256] fp32 (zero-padded per diag)
    const float* bias = (const float*)d_in[2];  // [511, 256] fp32
    float* out = (float*)d_out;                 // [128, 256, 256] fp32

    const size_t lds = (size_t)BDIM * (SDIM + XPAD) * sizeof(float); // 132,096 B (<320KB/WGP)
    dim3 grid(2 * SDIM - 1);                    // one workgroup per diagonal
    DiagonalTraining_wmma_kernel<<<grid, 256, lds, stream>>>(x, W, bias, out);
}